// wasserteinLoss_850403525440
// MI455X (gfx1250) — compile-verified
//
#include <hip/hip_runtime.h>
#include <math.h>

// Problem constants (compile-time, from the reference): B=64, H=W=512.
#define BATCH      64u
#define NLOG       18u
#define NSEG       (1u << NLOG)          // 262144 elements per array per batch (2^18)
#define TOTAL_ONE  (BATCH * NSEG)        // 2^24 floats per tensor
#define TOTAL_ALL  (2u * TOTAL_ONE)      // 2^25 floats combined (sA ++ sB)
#define TILE       2048u                 // LDS bitonic tile (8 KB of the 320 KB/WGP)
#define TTHREADS   1024u                 // threads per local-bitonic block (1 pair each)

typedef __attribute__((ext_vector_type(2))) float v2f;
typedef __attribute__((ext_vector_type(8))) float v8f;

// ---------------------------------------------------------------------------
// 1) Copy inputs into sortable workspace (vectorized b128 loads/stores).
// ---------------------------------------------------------------------------
__global__ __launch_bounds__(256) void copy_in(const float4* __restrict__ a,
                                               const float4* __restrict__ b,
                                               float4* __restrict__ sA,
                                               float4* __restrict__ sB) {
    unsigned i = blockIdx.x * blockDim.x + threadIdx.x;  // 2^22 float4 per tensor
    sA[i] = a[i];
    sB[i] = b[i];
}

// ---------------------------------------------------------------------------
// 2a) LDS-tiled bitonic stages. kBig==0: full local sort (k = 2..TILE).
//     kBig>0: finish stage kBig for j = TILE/2 .. 1 entirely in LDS.
//     Direction uses the segment-local index so all 128 segments of 2^18
//     sort ascending independently.
// ---------------------------------------------------------------------------
__global__ __launch_bounds__(TTHREADS) void bitonic_local(float* __restrict__ data,
                                                          unsigned kBig) {
    __shared__ float tile[TILE];
    const unsigned t    = threadIdx.x;
    const unsigned base = blockIdx.x * TILE;

    tile[t]            = data[base + t];
    tile[t + TTHREADS] = data[base + t + TTHREADS];
    __syncthreads();

    const unsigned segmask = NSEG - 1u;

    if (kBig == 0u) {
        for (unsigned k = 2u; k <= TILE; k <<= 1) {
            for (unsigned j = k >> 1; j >= 1u; j >>= 1) {
                unsigned il = ((t & ~(j - 1u)) << 1) | (t & (j - 1u));
                unsigned ir = il | j;
                unsigned li = (base + il) & segmask;
                bool up = ((li & k) == 0u);
                float x = tile[il], y = tile[ir];
                if (up ? (x > y) : (x < y)) { tile[il] = y; tile[ir] = x; }
                __syncthreads();
            }
        }
    } else {
        const unsigned k = kBig;
        for (unsigned j = TILE >> 1; j >= 1u; j >>= 1) {
            unsigned il = ((t & ~(j - 1u)) << 1) | (t & (j - 1u));
            unsigned ir = il | j;
            unsigned li = (base + il) & segmask;
            bool up = ((li & k) == 0u);
            float x = tile[il], y = tile[ir];
            if (up ? (x > y) : (x < y)) { tile[il] = y; tile[ir] = x; }
            __syncthreads();
        }
    }

    data[base + t]            = tile[t];
    data[base + t + TTHREADS] = tile[t + TTHREADS];
}

// ---------------------------------------------------------------------------
// 2b) Global bitonic compare-exchange for large strides (j >= TILE).
// ---------------------------------------------------------------------------
__global__ __launch_bounds__(256) void bitonic_global(float* __restrict__ data,
                                                      unsigned k, unsigned j) {
    unsigned t  = blockIdx.x * blockDim.x + threadIdx.x;   // 2^24 pairs
    unsigned i  = ((t & ~(j - 1u)) << 1) | (t & (j - 1u));
    unsigned l  = i | j;
    unsigned li = i & (NSEG - 1u);
    bool up = ((li & k) == 0u);
    float x = data[i], y = data[l];
    if (up ? (x > y) : (x < y)) { data[i] = y; data[l] = x; }
}

// ---------------------------------------------------------------------------
// 3) Per-element Cramer/W2 contributions via binary search into the other
//    sorted array. Exactly 64 deterministic partial sums per batch.
//    Grid: BATCH*64 blocks of 256 threads, 32 elements/thread (2^19 total/batch).
// ---------------------------------------------------------------------------
__global__ __launch_bounds__(256) void contrib(const float* __restrict__ sA,
                                               const float* __restrict__ sB,
                                               float* __restrict__ partial) {
    __shared__ float red[256];
    const unsigned b = blockIdx.x >> 6;   // batch
    const unsigned w = blockIdx.x & 63u;  // partial slot within batch
    const float* Av = sA + (size_t)b * NSEG;
    const float* Bv = sB + (size_t)b * NSEG;

    float acc = 0.f;
    for (unsigned it = 0; it < 32u; ++it) {
        unsigned e   = w * 8192u + it * 256u + threadIdx.x;   // [0, 2^19)
        unsigned isB = e >> NLOG;
        unsigned idx = e & (NSEG - 1u);
        const float* X = isB ? Bv : Av;   // element's own array
        const float* Y = isB ? Av : Bv;   // the other array
        float v = X[idx];
        __builtin_prefetch(X + idx + 256, 0, 1);   // gfx1250 global_prefetch_b8

        // upper_bound(Y, v): first index with Y[idx] > v  (18 probes, L2-resident)
        unsigned lo = 0u, hi = NSEG;
        while (lo < hi) {
            unsigned mid = (lo + hi) >> 1;
            float ym = Y[mid];
            if (ym <= v) lo = mid + 1u; else hi = mid;
        }

        int   di = (int)(idx + 1u) - (int)lo;      // (i1-i2); sign irrelevant (squared)
        float nX = (idx + 1u < NSEG) ? X[idx + 1u] : 3.0e38f;
        float nY = (lo < NSEG)       ? Y[lo]       : 3.0e38f;
        float nx = fminf(nX, nY);                  // next merged value
        if (nx < 1.0e38f) {                        // skip the global max (delta undefined)
            float d = (float)di;
            acc += d * d * (nx - v);
        }
    }

    // deterministic fixed-tree block reduction
    red[threadIdx.x] = acc;
    __syncthreads();
    for (unsigned s = 128u; s > 0u; s >>= 1) {
        if (threadIdx.x < s) red[threadIdx.x] += red[threadIdx.x + s];
        __syncthreads();
    }
    if (threadIdx.x == 0) partial[blockIdx.x] = red[0];   // P[b][w], row-major
}

// ---------------------------------------------------------------------------
// 4) Final: row-sums of P[64][64] on the WMMA unit, then sqrt + mean.
//    D(16x16,f32) = A(16x4,f32) x ones(4x16) + C  -> chained 16x along columns
//    gives exact row sums. A layout (ISA 7.12.2, 32-bit A 16x4):
//      lanes 0-15:  M=lane,    VGPR0=K0, VGPR1=K1
//      lanes 16-31: M=lane-16, VGPR0=K2, VGPR1=K3
//    D layout: lane L, vgpr v -> M = v + 8*(L>>4), N = L&15 (all N identical here).
//    One wave (32 threads) => EXEC all ones during every WMMA.
// ---------------------------------------------------------------------------
__global__ __launch_bounds__(32) void finalize(const float* __restrict__ partial,
                                               float* __restrict__ out) {
    __shared__ float sums[BATCH];
    const unsigned L = threadIdx.x;             // 0..31, single wave
    v2f ones; ones.x = 1.f; ones.y = 1.f;

    const unsigned row4 = L & 15u;              // M within 16-row tile
    const unsigned koff = (L >> 4) << 1;        // K pair base: 0 or 2

    for (unsigned rt = 0; rt < 4u; ++rt) {      // four 16-row tiles cover 64 batches
        v8f acc = {0.f, 0.f, 0.f, 0.f, 0.f, 0.f, 0.f, 0.f};
        unsigned row = rt * 16u + row4;
        for (unsigned tt = 0; tt < 16u; ++tt) { // 16 chained WMMAs sweep 64 columns
            unsigned col = tt * 4u + koff;
            v2f a;
            a.x = partial[row * 64u + col];
            a.y = partial[row * 64u + col + 1u];
            acc = __builtin_amdgcn_wmma_f32_16x16x4_f32(
                /*neg_a=*/false, a, /*neg_b=*/false, ones,
                /*c_mod=*/(short)0, acc, /*reuse_a=*/false, /*reuse_b=*/false);
        }
        if ((L & 15u) == 0u) {                  // lanes 0 and 16 own M = 0..7 / 8..15
            unsigned mbase = rt * 16u + ((L >> 4) << 3);
            for (int v = 0; v < 8; ++v) sums[mbase + v] = acc[v];
        }
    }
    __syncthreads();
    if (L == 0) {
        float m = 0.f;
        for (unsigned bb = 0; bb < BATCH; ++bb) m += sqrtf(sums[bb]);
        out[0] = m * (1.0f / ((float)BATCH * (float)NSEG));  // mean of sqrt(S)/n
    }
}

// ---------------------------------------------------------------------------
// Launch sequence (deterministic, all on `stream`, graph-capture safe).
// ws layout: sA[2^24] | sB[2^24] | partial[4096]  => ~128.02 MB required.
// ---------------------------------------------------------------------------
extern "C" void kernel_launch(void* const* d_in, const int* in_sizes, int n_in,
                              void* d_out, int out_size, void* d_ws, size_t ws_size,
                              hipStream_t stream) {
    const float* t1 = (const float*)d_in[0];
    const float* t2 = (const float*)d_in[1];
    float* sA      = (float*)d_ws;
    float* sB      = sA + TOTAL_ONE;
    float* partial = sB + TOTAL_ONE;   // 4096 floats
    float* outp    = (float*)d_out;

    // 1) stage inputs into workspace
    copy_in<<<TOTAL_ONE / (256u * 4u), 256, 0, stream>>>(
        (const float4*)t1, (const float4*)t2, (float4*)sA, (float4*)sB);

    // 2) segmented bitonic sort of 128 independent 2^18 segments (sA ++ sB contiguous)
    const unsigned nTiles = TOTAL_ALL / TILE;        // 16384 blocks
    const unsigned nPairB = (TOTAL_ALL / 2u) / 256u; // 65536 blocks
    bitonic_local<<<nTiles, TTHREADS, 0, stream>>>(sA, 0u);       // k = 2..2048 in LDS
    for (unsigned k = TILE * 2u; k <= NSEG; k <<= 1) {
        for (unsigned j = k >> 1; j >= TILE; j >>= 1)
            bitonic_global<<<nPairB, 256, 0, stream>>>(sA, k, j); // j = k/2 .. 2048
        bitonic_local<<<nTiles, TTHREADS, 0, stream>>>(sA, k);    // j = 1024..1 in LDS
    }

    // 3) contributions -> P[64][64] deterministic partials
    contrib<<<BATCH * 64u, 256, 0, stream>>>(sA, sB, partial);

    // 4) WMMA row-sum reduction + sqrt + mean
    finalize<<<1, 32, 0, stream>>>(partial, outp);
}